// DGCNN_Grouper_25735444037758
// MI455X (gfx1250) — compile-verified
//
#include <hip/hip_runtime.h>

typedef float v2f __attribute__((ext_vector_type(2)));
typedef float v8f __attribute__((ext_vector_type(8)));

#define CEIL(a, b) (((a) + (b) - 1) / (b))
#define KNNK 16

// ---------------------------------------------------------------------------
// f0 = Wt (8x3) @ x^T + bt  ->  (B, 8, N)
// ---------------------------------------------------------------------------
__global__ void lift_kernel(const float* __restrict__ x, const float* __restrict__ Wt,
                            const float* __restrict__ bt, float* __restrict__ f0,
                            int B, int N) {
  int gid = blockIdx.x * blockDim.x + threadIdx.x;
  if (gid >= B * N) return;
  int n = gid % N, b = gid / N;
  const float* xp = x + (size_t)gid * 3;
  float x0 = xp[0], x1 = xp[1], x2 = xp[2];
#pragma unroll
  for (int o = 0; o < 8; ++o) {
    f0[((size_t)b * 8 + o) * N + n] =
        Wt[o * 3 + 0] * x0 + Wt[o * 3 + 1] * x1 + Wt[o * 3 + 2] * x2 + bt[o];
  }
}

// ---------------------------------------------------------------------------
// Brute-force KNN: for each query q in cq find 16 nearest points in ck.
// Stable insertion (strict <) matches jax.lax.top_k tie-break (lower index).
// ---------------------------------------------------------------------------
__global__ void knn_kernel(const float* __restrict__ cq, const float* __restrict__ ck,
                           int B, int Nq, int Nk, int* __restrict__ idx) {
  int gid = blockIdx.x * blockDim.x + threadIdx.x;
  if (gid >= B * Nq) return;
  int q = gid % Nq, b = gid / Nq;
  const float* cqb = cq + ((size_t)b * Nq + q) * 3;
  const float* ckb = ck + (size_t)b * Nk * 3;
  float qx = cqb[0], qy = cqb[1], qz = cqb[2];
  float qn = qx * qx + qy * qy + qz * qz;

  float bd[KNNK];
  int   bi[KNNK];
#pragma unroll
  for (int t = 0; t < KNNK; ++t) { bd[t] = 3.4e38f; bi[t] = 0; }

  for (int j = 0; j < Nk; ++j) {
    float kx = ckb[j * 3 + 0], ky = ckb[j * 3 + 1], kz = ckb[j * 3 + 2];
    float d = qn + kx * kx + ky * ky + kz * kz - 2.0f * (qx * kx + qy * ky + qz * kz);
    if (d < bd[KNNK - 1]) {
#pragma unroll
      for (int t = KNNK - 1; t > 0; --t) {
        float pd = bd[t - 1];
        bool shift = d < pd;            // element at t-1 moves right
        bool here  = (!shift) && (d < bd[t]);  // d lands at slot t
        bd[t] = shift ? pd        : (here ? d : bd[t]);
        bi[t] = shift ? bi[t - 1] : (here ? j : bi[t]);
      }
      if (d < bd[0]) { bd[0] = d; bi[0] = j; }
    }
  }
  int* op = idx + (size_t)gid * KNNK;
#pragma unroll
  for (int t = 0; t < KNNK; ++t) op[t] = bi[t];
}

// ---------------------------------------------------------------------------
// Graph feature: out (B, 2C, Nq*16); ch [0,C): xk[idx]-xq ; ch [C,2C): xq
// ---------------------------------------------------------------------------
__global__ void build_graph_kernel(const float* __restrict__ xq, const float* __restrict__ xk,
                                   const int* __restrict__ idx, float* __restrict__ out,
                                   int B, int C, int Nq, int Nk) {
  int gid = blockIdx.x * blockDim.x + threadIdx.x;
  if (gid >= B * Nq * KNNK) return;
  int kk = gid & (KNNK - 1);
  int q = (gid / KNNK) % Nq;
  int b = gid / (Nq * KNNK);
  int j = idx[gid];
  const float* xqb = xq + (size_t)b * C * Nq;
  const float* xkb = xk + (size_t)b * C * Nk;
  float* ob = out + (size_t)b * (2 * C) * ((size_t)Nq * KNNK);
  size_t col = (size_t)q * KNNK + kk;
  size_t ld = (size_t)Nq * KNNK;
  for (int c = 0; c < C; ++c) {
    float fq = xqb[(size_t)c * Nq + q];
    float fn = xkb[(size_t)c * Nk + j];
    ob[(size_t)c * ld + col] = fn - fq;
    ob[(size_t)(C + c) * ld + col] = fq;
  }
}

// ---------------------------------------------------------------------------
// WMMA GEMM: Y[b] = W (O x I) @ F[b] (I x NK), f32 via V_WMMA_F32_16X16X4_F32
// One wave32 per 16x16 output tile. O%16==0, I%4==0, NK%16==0 guaranteed.
// A frag (16x4 f32, ISA layout): lanes0-15 hold K=i0,i0+1; lanes16-31 K=i0+2,i0+3.
// C/D layout: VGPR r -> M=r (lanes0-15), M=r+8 (lanes16-31); N = lane&15.
// ---------------------------------------------------------------------------
__global__ void wmma_gemm_kernel(const float* __restrict__ W, const float* __restrict__ F,
                                 float* __restrict__ Y, int O, int I, int NK) {
  int tile_n = blockIdx.x;   // NK / 16
  int tile_o = blockIdx.y;   // O  / 16
  int b = blockIdx.z;
  int lane = threadIdx.x;    // 0..31
  int m = lane & 15;
  int half = lane >> 4;      // 0 or 1

  const float* Fb = F + (size_t)b * I * NK;
  float* Yb = Y + (size_t)b * O * NK;
  int col = tile_n * 16 + m;
  int orow = tile_o * 16 + m;

  v8f acc = {0.f, 0.f, 0.f, 0.f, 0.f, 0.f, 0.f, 0.f};
  for (int i0 = 0; i0 < I; i0 += 4) {
    v2f a, bf;
    const float* wr = W + (size_t)orow * I + i0 + half * 2;
    a.x = wr[0];
    a.y = wr[1];
    const float* fr = Fb + (size_t)(i0 + half * 2) * NK + col;
    bf.x = fr[0];
    bf.y = fr[(size_t)NK];
    acc = __builtin_amdgcn_wmma_f32_16x16x4_f32(false, a, false, bf,
                                                (short)0, acc, false, false);
  }
#pragma unroll
  for (int r = 0; r < 8; ++r) {
    int row = tile_o * 16 + r + half * 8;
    Yb[(size_t)row * NK + col] = acc[r];
  }
}

// ---------------------------------------------------------------------------
// GroupNorm stats: one block per (b, g); mean & rstd over (O/G)*NK elements
// ---------------------------------------------------------------------------
__global__ void gn_stats_kernel(const float* __restrict__ Y, float* __restrict__ stats,
                                int O, int NK, int G) {
  int bg = blockIdx.x;
  int g = bg % G, b = bg / G;
  int cg = O / G;
  size_t n = (size_t)cg * NK;
  const float* base = Y + ((size_t)b * O + (size_t)g * cg) * NK;
  int tid = threadIdx.x;
  float s = 0.f, sq = 0.f;
  for (size_t i = tid; i < n; i += blockDim.x) {
    float v = base[i];
    s += v;
    sq += v * v;
  }
  __shared__ float shs[256];
  __shared__ float shq[256];
  shs[tid] = s; shq[tid] = sq;
  __syncthreads();
  for (int st = 128; st > 0; st >>= 1) {
    if (tid < st) { shs[tid] += shs[tid + st]; shq[tid] += shq[tid + st]; }
    __syncthreads();
  }
  if (tid == 0) {
    float inv = 1.0f / (float)n;
    float mu = shs[0] * inv;
    float var = shq[0] * inv - mu * mu;
    stats[bg * 2 + 0] = mu;
    stats[bg * 2 + 1] = rsqrtf(var + 1e-5f);
  }
}

// ---------------------------------------------------------------------------
// normalize + affine + LeakyReLU + max over k  ->  out (B, O, Nq)
// ---------------------------------------------------------------------------
__global__ void gn_act_max_kernel(const float* __restrict__ Y, const float* __restrict__ stats,
                                  const float* __restrict__ gamma, const float* __restrict__ beta,
                                  float* __restrict__ out, int B, int O, int Nq, int G) {
  int gid = blockIdx.x * blockDim.x + threadIdx.x;
  if (gid >= B * O * Nq) return;
  int n = gid % Nq;
  int o = (gid / Nq) % O;
  int b = gid / (O * Nq);
  int cg = O / G;
  int g = o / cg;
  float mu = stats[(b * G + g) * 2 + 0];
  float rs = stats[(b * G + g) * 2 + 1];
  float ga = gamma[o], be = beta[o];
  const float* yp = Y + ((size_t)b * O + o) * ((size_t)Nq * KNNK) + (size_t)n * KNNK;
  float mx = -3.4e38f;
#pragma unroll
  for (int kk = 0; kk < KNNK; ++kk) {
    float v = (yp[kk] - mu) * rs * ga + be;
    v = (v >= 0.f) ? v : 0.2f * v;
    mx = fmaxf(mx, v);
  }
  out[gid] = mx;
}

// ---------------------------------------------------------------------------
// Farthest point sampling: one 256-thread block per batch; dists in LDS.
// Matches reference scan: outputs far-in (starting at 0), argmax afterwards,
// argmax tie -> lowest index (jnp.argmax semantics).
// ---------------------------------------------------------------------------
__global__ void fps_kernel(const float* __restrict__ coor, int N, int num,
                           int* __restrict__ outidx) {
  int b = blockIdx.x;
  const float* c = coor + (size_t)b * N * 3;
  __shared__ float dists[2048];
  __shared__ float rv[256];
  __shared__ int ri[256];
  __shared__ int s_far;
  int tid = threadIdx.x;
  for (int j = tid; j < N; j += 256) dists[j] = 1e10f;
  if (tid == 0) s_far = 0;
  __syncthreads();
  for (int t = 0; t < num; ++t) {
    int far = s_far;
    if (tid == 0) outidx[(size_t)b * num + t] = far;
    float fx = c[far * 3 + 0], fy = c[far * 3 + 1], fz = c[far * 3 + 2];
    float bv = -1.f;
    int bidx = 0x7fffffff;
    for (int j = tid; j < N; j += 256) {
      float dx = c[j * 3 + 0] - fx;
      float dy = c[j * 3 + 1] - fy;
      float dz = c[j * 3 + 2] - fz;
      float d = dx * dx + dy * dy + dz * dz;
      float nd = fminf(dists[j], d);
      dists[j] = nd;
      if (nd > bv) { bv = nd; bidx = j; }  // ascending j keeps first max
    }
    rv[tid] = bv; ri[tid] = bidx;
    __syncthreads();
    for (int st = 128; st > 0; st >>= 1) {
      if (tid < st) {
        float ov = rv[tid + st]; int oi = ri[tid + st];
        if (ov > rv[tid] || (ov == rv[tid] && oi < ri[tid])) { rv[tid] = ov; ri[tid] = oi; }
      }
      __syncthreads();
    }
    if (tid == 0) s_far = ri[0];
    __syncthreads();
  }
}

// ---------------------------------------------------------------------------
// Gathers for fps_downsample
// ---------------------------------------------------------------------------
__global__ void gather_coor_kernel(const float* __restrict__ coor, const int* __restrict__ idx,
                                   float* __restrict__ out, int B, int N, int num) {
  int gid = blockIdx.x * blockDim.x + threadIdx.x;
  if (gid >= B * num) return;
  int q = gid % num, b = gid / num;
  int j = idx[(size_t)b * num + q];
  const float* src = coor + ((size_t)b * N + j) * 3;
  float* dst = out + (size_t)gid * 3;
  dst[0] = src[0]; dst[1] = src[1]; dst[2] = src[2];
}

__global__ void gather_feat_kernel(const float* __restrict__ f, const int* __restrict__ idx,
                                   float* __restrict__ out, int B, int C, int N, int num) {
  int gid = blockIdx.x * blockDim.x + threadIdx.x;
  if (gid >= B * C * num) return;
  int q = gid % num;
  int c = (gid / num) % C;
  int b = gid / (C * num);
  out[gid] = f[((size_t)b * C + c) * N + idx[(size_t)b * num + q]];
}

// encoded = swapaxes(s3, 1, 2): out[b][n][c] = s[b][c][n]
__global__ void transpose_cn_kernel(const float* __restrict__ s, float* __restrict__ out,
                                    int B, int C, int N) {
  int gid = blockIdx.x * blockDim.x + threadIdx.x;
  if (gid >= B * C * N) return;
  int c = gid % C;
  int n = (gid / C) % N;
  int b = gid / (C * N);
  out[gid] = s[((size_t)b * C + c) * N + n];
}

// ---------------------------------------------------------------------------
// Host orchestration
// ---------------------------------------------------------------------------
extern "C" void kernel_launch(void* const* d_in, const int* in_sizes, int n_in,
                              void* d_out, int out_size, void* d_ws, size_t ws_size,
                              hipStream_t stream) {
  (void)in_sizes; (void)n_in; (void)out_size; (void)ws_size;
  const int B = 16, N0 = 2048, NUM0 = 512, NUM1 = 256, NUM2 = 128, G = 4;

  const float* x  = (const float*)d_in[0];
  const float* Wt = (const float*)d_in[4];
  const float* bt = (const float*)d_in[5];
  const float* W1 = (const float*)d_in[6];
  const float* g1 = (const float*)d_in[7];
  const float* b1 = (const float*)d_in[8];
  const float* W2 = (const float*)d_in[9];
  const float* g2 = (const float*)d_in[10];
  const float* b2 = (const float*)d_in[11];
  const float* W3 = (const float*)d_in[12];
  const float* g3 = (const float*)d_in[13];
  const float* b3 = (const float*)d_in[14];
  const float* W4 = (const float*)d_in[15];
  const float* g4 = (const float*)d_in[16];
  const float* b4 = (const float*)d_in[17];
  const float* W5 = (const float*)d_in[18];
  const float* g5 = (const float*)d_in[19];
  const float* b5 = (const float*)d_in[20];
  const float* W6 = (const float*)d_in[21];
  const float* g6 = (const float*)d_in[22];
  const float* b6 = (const float*)d_in[23];
  const float* W7 = (const float*)d_in[24];
  const float* g7 = (const float*)d_in[25];
  const float* b7 = (const float*)d_in[26];

  // d_out regions: s1, s2, s3, p1, p2, p3, encoded
  float* out = (float*)d_out;
  float* S1 = out;                         // (B, 64, 512)
  float* S2 = S1 + (size_t)B * 64 * 512;   // (B, 128, 256)
  float* S3 = S2 + (size_t)B * 128 * 256;  // (B, 256, 128)
  float* P1 = S3 + (size_t)B * 256 * 128;  // (B, 512, 3)
  float* P2 = P1 + (size_t)B * 512 * 3;    // (B, 256, 3)
  float* P3 = P2 + (size_t)B * 256 * 3;    // (B, 128, 3)
  float* ENC = P3 + (size_t)B * 128 * 3;   // (B, 128, 256)

  // Workspace carve-up
  float* wsf = (float*)d_ws;
  size_t off = 0;
  float* F0 = wsf + off; off += (size_t)B * 8 * 2048;        // lifted features
  float* FA = wsf + off; off += (size_t)B * 512 * 128 * 16;  // graph feature (max 16.78M f)
  float* FY = wsf + off; off += (size_t)B * 32 * 2048 * 16;  // GEMM out (max 16.78M f)
  float* F1 = wsf + off; off += (size_t)B * 32 * 2048;
  float* FQ = wsf + off; off += (size_t)B * 128 * 512;
  float* F2 = wsf + off; off += (size_t)B * 64 * 512;
  float* F4 = wsf + off; off += (size_t)B * 128 * 256;
  float* F6 = wsf + off; off += (size_t)B * 256 * 128;
  float* STATS = wsf + off; off += (size_t)B * G * 2;
  int* IDX  = (int*)(wsf + off); off += (size_t)B * 2048 * 16;
  int* FPSI = (int*)(wsf + off); off += (size_t)B * 512;

  // --- Stage 0: lift ---
  lift_kernel<<<CEIL(B * N0, 256), 256, 0, stream>>>(x, Wt, bt, F0, B, N0);

  // --- Stage 1: graph(coor,f0 | coor,f0) -> conv W1 -> F1 (B,32,2048) ---
  knn_kernel<<<CEIL(B * N0, 64), 64, 0, stream>>>(x, x, B, N0, N0, IDX);
  build_graph_kernel<<<CEIL(B * N0 * 16, 256), 256, 0, stream>>>(F0, F0, IDX, FA, B, 8, N0, N0);
  wmma_gemm_kernel<<<dim3(N0 * 16 / 16, 32 / 16, B), 32, 0, stream>>>(W1, FA, FY, 32, 16, N0 * 16);
  gn_stats_kernel<<<B * G, 256, 0, stream>>>(FY, STATS, 32, N0 * 16, G);
  gn_act_max_kernel<<<CEIL(B * 32 * N0, 256), 256, 0, stream>>>(FY, STATS, g1, b1, F1, B, 32, N0, G);

  // --- FPS to 512 ---
  fps_kernel<<<B, 256, 0, stream>>>(x, N0, NUM0, FPSI);
  gather_coor_kernel<<<CEIL(B * NUM0, 256), 256, 0, stream>>>(x, FPSI, P1, B, N0, NUM0);
  gather_feat_kernel<<<CEIL(B * 32 * NUM0, 256), 256, 0, stream>>>(F1, FPSI, FQ, B, 32, N0, NUM0);

  // --- Stage 2: graph(p1,FQ | coor,F1) -> conv W2 -> F2 (B,64,512) ---
  knn_kernel<<<CEIL(B * NUM0, 64), 64, 0, stream>>>(P1, x, B, NUM0, N0, IDX);
  build_graph_kernel<<<CEIL(B * NUM0 * 16, 256), 256, 0, stream>>>(FQ, F1, IDX, FA, B, 32, NUM0, N0);
  wmma_gemm_kernel<<<dim3(NUM0 * 16 / 16, 64 / 16, B), 32, 0, stream>>>(W2, FA, FY, 64, 64, NUM0 * 16);
  gn_stats_kernel<<<B * G, 256, 0, stream>>>(FY, STATS, 64, NUM0 * 16, G);
  gn_act_max_kernel<<<CEIL(B * 64 * NUM0, 256), 256, 0, stream>>>(FY, STATS, g2, b2, F2, B, 64, NUM0, G);

  // --- Stage 3: graph(p1,F2 | p1,F2) -> conv W3 -> s1 (B,64,512) ---
  knn_kernel<<<CEIL(B * NUM0, 64), 64, 0, stream>>>(P1, P1, B, NUM0, NUM0, IDX);
  build_graph_kernel<<<CEIL(B * NUM0 * 16, 256), 256, 0, stream>>>(F2, F2, IDX, FA, B, 64, NUM0, NUM0);
  wmma_gemm_kernel<<<dim3(NUM0 * 16 / 16, 64 / 16, B), 32, 0, stream>>>(W3, FA, FY, 64, 128, NUM0 * 16);
  gn_stats_kernel<<<B * G, 256, 0, stream>>>(FY, STATS, 64, NUM0 * 16, G);
  gn_act_max_kernel<<<CEIL(B * 64 * NUM0, 256), 256, 0, stream>>>(FY, STATS, g3, b3, S1, B, 64, NUM0, G);

  // --- FPS to 256 ---
  fps_kernel<<<B, 256, 0, stream>>>(P1, NUM0, NUM1, FPSI);
  gather_coor_kernel<<<CEIL(B * NUM1, 256), 256, 0, stream>>>(P1, FPSI, P2, B, NUM0, NUM1);
  gather_feat_kernel<<<CEIL(B * 64 * NUM1, 256), 256, 0, stream>>>(S1, FPSI, FQ, B, 64, NUM0, NUM1);

  // --- Stage 4: graph(p2,FQ | p1,s1) -> conv W4 -> F4 (B,128,256) ---
  knn_kernel<<<CEIL(B * NUM1, 64), 64, 0, stream>>>(P2, P1, B, NUM1, NUM0, IDX);
  build_graph_kernel<<<CEIL(B * NUM1 * 16, 256), 256, 0, stream>>>(FQ, S1, IDX, FA, B, 64, NUM1, NUM0);
  wmma_gemm_kernel<<<dim3(NUM1 * 16 / 16, 128 / 16, B), 32, 0, stream>>>(W4, FA, FY, 128, 128, NUM1 * 16);
  gn_stats_kernel<<<B * G, 256, 0, stream>>>(FY, STATS, 128, NUM1 * 16, G);
  gn_act_max_kernel<<<CEIL(B * 128 * NUM1, 256), 256, 0, stream>>>(FY, STATS, g4, b4, F4, B, 128, NUM1, G);

  // --- Stage 5: graph(p2,F4 | p2,F4) -> conv W5 -> s2 (B,128,256) ---
  knn_kernel<<<CEIL(B * NUM1, 64), 64, 0, stream>>>(P2, P2, B, NUM1, NUM1, IDX);
  build_graph_kernel<<<CEIL(B * NUM1 * 16, 256), 256, 0, stream>>>(F4, F4, IDX, FA, B, 128, NUM1, NUM1);
  wmma_gemm_kernel<<<dim3(NUM1 * 16 / 16, 128 / 16, B), 32, 0, stream>>>(W5, FA, FY, 128, 256, NUM1 * 16);
  gn_stats_kernel<<<B * G, 256, 0, stream>>>(FY, STATS, 128, NUM1 * 16, G);
  gn_act_max_kernel<<<CEIL(B * 128 * NUM1, 256), 256, 0, stream>>>(FY, STATS, g5, b5, S2, B, 128, NUM1, G);

  // --- FPS to 128 ---
  fps_kernel<<<B, 256, 0, stream>>>(P2, NUM1, NUM2, FPSI);
  gather_coor_kernel<<<CEIL(B * NUM2, 256), 256, 0, stream>>>(P2, FPSI, P3, B, NUM1, NUM2);
  gather_feat_kernel<<<CEIL(B * 128 * NUM2, 256), 256, 0, stream>>>(S2, FPSI, FQ, B, 128, NUM1, NUM2);

  // --- Stage 6: graph(p3,FQ | p2,s2) -> conv W6 -> F6 (B,256,128) ---
  knn_kernel<<<CEIL(B * NUM2, 64), 64, 0, stream>>>(P3, P2, B, NUM2, NUM1, IDX);
  build_graph_kernel<<<CEIL(B * NUM2 * 16, 256), 256, 0, stream>>>(FQ, S2, IDX, FA, B, 128, NUM2, NUM1);
  wmma_gemm_kernel<<<dim3(NUM2 * 16 / 16, 256 / 16, B), 32, 0, stream>>>(W6, FA, FY, 256, 256, NUM2 * 16);
  gn_stats_kernel<<<B * G, 256, 0, stream>>>(FY, STATS, 256, NUM2 * 16, G);
  gn_act_max_kernel<<<CEIL(B * 256 * NUM2, 256), 256, 0, stream>>>(FY, STATS, g6, b6, F6, B, 256, NUM2, G);

  // --- Stage 7: graph(p3,F6 | p3,F6) -> conv W7 -> s3 (B,256,128) ---
  knn_kernel<<<CEIL(B * NUM2, 64), 64, 0, stream>>>(P3, P3, B, NUM2, NUM2, IDX);
  build_graph_kernel<<<CEIL(B * NUM2 * 16, 256), 256, 0, stream>>>(F6, F6, IDX, FA, B, 256, NUM2, NUM2);
  wmma_gemm_kernel<<<dim3(NUM2 * 16 / 16, 256 / 16, B), 32, 0, stream>>>(W7, FA, FY, 256, 512, NUM2 * 16);
  gn_stats_kernel<<<B * G, 256, 0, stream>>>(FY, STATS, 256, NUM2 * 16, G);
  gn_act_max_kernel<<<CEIL(B * 256 * NUM2, 256), 256, 0, stream>>>(FY, STATS, g7, b7, S3, B, 256, NUM2, G);

  // --- encoded = swapaxes(s3, 1, 2) ---
  transpose_cn_kernel<<<CEIL(B * 256 * 128, 256), 256, 0, stream>>>(S3, ENC, B, 256, NUM2);
}